// GATEncoder_40553081209345
// MI455X (gfx1250) — compile-verified
//
#include <hip/hip_runtime.h>
#include <math.h>

typedef __attribute__((ext_vector_type(2))) float v2f;
typedef __attribute__((ext_vector_type(8))) float v8f;

#define NEG_SLOPE 0.2f

__device__ __forceinline__ float lrelu(float v) {
    return v > 0.0f ? v : NEG_SLOPE * v;
}

// float atomic max via ordered-int trick (positive floats compare as signed ints,
// negative floats compare inverted as unsigned ints).
__device__ __forceinline__ void atomicMaxF(float* addr, float val) {
    if (val >= 0.0f) {
        atomicMax((int*)addr, __float_as_int(val));
    } else {
        atomicMin((unsigned int*)addr, __float_as_uint(val));
    }
}

// ---------------------------------------------------------------------------
// 0) workspace init (amax=-inf, asum=0, accum=0, meansum=0) — every call
// ---------------------------------------------------------------------------
__global__ void init_ws(float* amax, float* asum, float* accum, float* meansum, int N) {
    int gid = blockIdx.x * blockDim.x + threadIdx.x;
    int stride = gridDim.x * blockDim.x;
    for (int i = gid; i < N * 64; i += stride) accum[i] = 0.0f;
    for (int i = gid; i < N * 4; i += stride) {
        amax[i] = -INFINITY;
        asum[i] = 0.0f;
    }
    if (gid == 0) meansum[0] = 0.0f;
}

// ---------------------------------------------------------------------------
// 1) mean(edge_attr) partial reduction
// ---------------------------------------------------------------------------
__global__ void mean_reduce(const float* __restrict__ ea, float* meansum, int E) {
    __shared__ float sm[256];
    int t = threadIdx.x;
    float s = 0.0f;
    for (int i = blockIdx.x * blockDim.x + t; i < E; i += gridDim.x * blockDim.x)
        s += ea[i];
    sm[t] = s;
    __syncthreads();
    for (int off = 128; off >= 1; off >>= 1) {
        if (t < off) sm[t] += sm[t + off];
        __syncthreads();
    }
    if (t == 0) atomicAdd(meansum, sm[0]);
}

// ---------------------------------------------------------------------------
// 2) coef[h] = dot(lin_edge_w[h*64: ], att_edge[h*64: ]) ; finalize mean
// ---------------------------------------------------------------------------
__global__ void coef_kernel(const float* __restrict__ lin, const float* __restrict__ att_e,
                            const float* __restrict__ meansum, float* meanv, float* coef, int E) {
    __shared__ float sm[256];
    int t = threadIdx.x;
    sm[t] = lin[t] * att_e[t];
    __syncthreads();
    for (int off = 32; off >= 1; off >>= 1) {
        if ((t & 63) < off) sm[t] += sm[t + off];
        __syncthreads();
    }
    if ((t & 63) == 0) coef[t >> 6] = sm[t];
    if (t == 0) meanv[0] = meansum[0] / (float)E;
}

// ---------------------------------------------------------------------------
// 3) h = x @ W  via V_WMMA_F32_16X16X4_F32
//    block = 256 threads = 8 waves; block covers 16 rows x 256 cols,
//    wave w covers 2 column tiles (cols 32w..32w+31). K=64 in 16 steps.
// ---------------------------------------------------------------------------
__global__ void gemm_xw(const float* __restrict__ x, const float* __restrict__ W,
                        float* __restrict__ hbuf, int N) {
    const int lane = threadIdx.x & 31;
    const int wave = threadIdx.x >> 5;
    const int row0 = blockIdx.x * 16;
    const int m    = lane & 15;          // A row / B,C col within tile
    const int kp   = (lane >> 4) << 1;   // 0 or 2: K sub-pair held by this lane
    const int col0 = wave * 32;

    int rload = row0 + m;
    if (rload > N - 1) rload = N - 1;    // clamp (EXEC must stay all-ones for WMMA)
    const float* xr = x + (size_t)rload * 64 + kp;

    v8f acc0 = {};
    v8f acc1 = {};

#pragma unroll
    for (int k0 = 0; k0 < 64; k0 += 4) {
        // A fragment 16x4: lane<16 holds (K=k0, k0+1), lane>=16 holds (k0+2, k0+3)
        float2 av = *(const float2*)(xr + k0);
        v2f a;
        a[0] = av.x;
        a[1] = av.y;
        // B fragments 4x16 for the two column tiles (row-striped across lanes)
        const float* wp = W + (size_t)(k0 + kp) * 256 + col0 + m;
        v2f b0, b1;
        b0[0] = wp[0];
        b0[1] = wp[256];
        b1[0] = wp[16];
        b1[1] = wp[256 + 16];
        // (neg_a, A, neg_b, B, c_mod, C, reuse_a, reuse_b)
        acc0 = __builtin_amdgcn_wmma_f32_16x16x4_f32(false, a, false, b0, (short)0, acc0,
                                                     false, false);
        acc1 = __builtin_amdgcn_wmma_f32_16x16x4_f32(false, a, false, b1, (short)0, acc1,
                                                     false, false);
    }

    // C/D layout: VGPR i -> M = i (lanes 0-15) / M = 8+i (lanes 16-31), N = lane&15
    const int rbase = row0 + ((lane >> 4) << 3);
    const int cw = col0 + m;
#pragma unroll
    for (int i = 0; i < 8; ++i) {
        int r = rbase + i;
        if (r < N) {
            hbuf[(size_t)r * 256 + cw]      = acc0[i];
            hbuf[(size_t)r * 256 + cw + 16] = acc1[i];
        }
    }
}

// ---------------------------------------------------------------------------
// 4) per-(node, head) attention scalars a_src, a_dst
// ---------------------------------------------------------------------------
__global__ void node_att(const float* __restrict__ hbuf, const float* __restrict__ att_src,
                         const float* __restrict__ att_dst, float* asrc, float* adst, int N) {
    int gid = blockIdx.x * blockDim.x + threadIdx.x;
    if (gid >= N * 4) return;
    int n = gid >> 2, hd = gid & 3;
    const float* hp = hbuf + (size_t)n * 256 + hd * 64;
    const float* as = att_src + hd * 64;
    const float* ad = att_dst + hd * 64;
    float s1 = 0.0f, s2 = 0.0f;
#pragma unroll 4
    for (int c = 0; c < 64; ++c) {
        float v = hp[c];
        s1 += v * as[c];
        s2 += v * ad[c];
    }
    asrc[gid] = s1;
    adst[gid] = s2;
}

// ---------------------------------------------------------------------------
// 5) per-edge logits -> leaky-relu -> segment max over dst
// ---------------------------------------------------------------------------
__global__ void edge_max(const int* __restrict__ ei, const float* __restrict__ eattr,
                         const float* __restrict__ asrc, const float* __restrict__ adst,
                         const float* __restrict__ coef, const float* __restrict__ meanv,
                         float* amax, int E, int N) {
    int e = blockIdx.x * blockDim.x + threadIdx.x;
    if (e >= E + N) return;
    int s, d;
    float ea;
    if (e < E) { s = ei[e]; d = ei[E + e]; ea = eattr[e]; }
    else       { s = e - E; d = s;         ea = meanv[0]; }
#pragma unroll
    for (int h = 0; h < 4; ++h) {
        float l = lrelu(asrc[s * 4 + h] + adst[d * 4 + h] + ea * coef[h]);
        atomicMaxF(&amax[d * 4 + h], l);
    }
}

// ---------------------------------------------------------------------------
// 6) exp(logit - max) -> segment sum over dst (logit recomputed; all L2-hot)
// ---------------------------------------------------------------------------
__global__ void edge_expsum(const int* __restrict__ ei, const float* __restrict__ eattr,
                            const float* __restrict__ asrc, const float* __restrict__ adst,
                            const float* __restrict__ coef, const float* __restrict__ meanv,
                            const float* __restrict__ amax, float* asum, int E, int N) {
    int e = blockIdx.x * blockDim.x + threadIdx.x;
    if (e >= E + N) return;
    int s, d;
    float ea;
    if (e < E) { s = ei[e]; d = ei[E + e]; ea = eattr[e]; }
    else       { s = e - E; d = s;         ea = meanv[0]; }
#pragma unroll
    for (int h = 0; h < 4; ++h) {
        float l = lrelu(asrc[s * 4 + h] + adst[d * 4 + h] + ea * coef[h]);
        atomicAdd(&asum[d * 4 + h], __expf(l - amax[d * 4 + h]));
    }
}

// ---------------------------------------------------------------------------
// 7) weighted scatter: 64 threads per edge; fold the 4-head mean in registers
//    -> one atomicAdd per (edge, channel). h and accum are L2-resident.
// ---------------------------------------------------------------------------
__global__ void edge_scatter(const int* __restrict__ ei, const float* __restrict__ eattr,
                             const float* __restrict__ asrc, const float* __restrict__ adst,
                             const float* __restrict__ coef, const float* __restrict__ meanv,
                             const float* __restrict__ amax, const float* __restrict__ asum,
                             const float* __restrict__ hbuf, float* accum, int E, int N) {
    long long gid = (long long)blockIdx.x * blockDim.x + threadIdx.x;
    int e = (int)(gid >> 6);
    int c = (int)(gid & 63);
    if (e >= E + N) return;
    int s, d;
    float ea;
    if (e < E) { s = ei[e]; d = ei[E + e]; ea = eattr[e]; }
    else       { s = e - E; d = s;         ea = meanv[0]; }

    const float* hrow = hbuf + (size_t)s * 256;
    __builtin_prefetch(hrow + c, 0, 1);  // global_prefetch_b8

    float acc = 0.0f;
#pragma unroll
    for (int h = 0; h < 4; ++h) {
        float l = lrelu(asrc[s * 4 + h] + adst[d * 4 + h] + ea * coef[h]);
        float w = __expf(l - amax[d * 4 + h]) / (asum[d * 4 + h] + 1e-16f);
        acc += w * hrow[h * 64 + c];
    }
    atomicAdd(&accum[(size_t)d * 64 + c], 0.25f * acc);  // head mean folded in
}

// ---------------------------------------------------------------------------
// 8) finalize: relu(mean + bias) + x
// ---------------------------------------------------------------------------
__global__ void finalize(const float* __restrict__ accum, const float* __restrict__ bias,
                         const float* __restrict__ x, float* __restrict__ out, int N) {
    int gid = blockIdx.x * blockDim.x + threadIdx.x;
    if (gid >= N * 64) return;
    int c = gid & 63;
    float v = accum[gid] + bias[c];
    out[gid] = fmaxf(v, 0.0f) + x[gid];
}

extern "C" void kernel_launch(void* const* d_in, const int* in_sizes, int n_in,
                              void* d_out, int out_size, void* d_ws, size_t ws_size,
                              hipStream_t stream) {
    const float* x        = (const float*)d_in[0];
    const int*   ei       = (const int*)d_in[1];
    const float* eattr    = (const float*)d_in[2];
    const float* W        = (const float*)d_in[3];
    const float* att_src  = (const float*)d_in[4];
    const float* att_dst  = (const float*)d_in[5];
    const float* lin_edge = (const float*)d_in[6];
    const float* att_edge = (const float*)d_in[7];
    const float* bias     = (const float*)d_in[8];
    float* out = (float*)d_out;

    const int N   = in_sizes[0] / 64;
    const int E   = in_sizes[1] / 2;
    const int TOT = E + N;

    float* ws      = (float*)d_ws;
    float* hbuf    = ws;                          // N*256
    float* accum   = hbuf  + (size_t)N * 256;     // N*64
    float* asrc    = accum + (size_t)N * 64;      // N*4
    float* adst    = asrc  + (size_t)N * 4;       // N*4
    float* amax    = adst  + (size_t)N * 4;       // N*4
    float* asum    = amax  + (size_t)N * 4;       // N*4
    float* meansum = asum  + (size_t)N * 4;       // 1
    float* meanv   = meansum + 1;                 // 1
    float* coef    = meanv + 1;                   // 4

    init_ws<<<1024, 256, 0, stream>>>(amax, asum, accum, meansum, N);
    mean_reduce<<<512, 256, 0, stream>>>(eattr, meansum, E);
    coef_kernel<<<1, 256, 0, stream>>>(lin_edge, att_edge, meansum, meanv, coef, E);
    gemm_xw<<<(N + 15) / 16, 256, 0, stream>>>(x, W, hbuf, N);
    node_att<<<(N * 4 + 255) / 256, 256, 0, stream>>>(hbuf, att_src, att_dst, asrc, adst, N);
    edge_max<<<(TOT + 255) / 256, 256, 0, stream>>>(ei, eattr, asrc, adst, coef, meanv, amax, E, N);
    edge_expsum<<<(TOT + 255) / 256, 256, 0, stream>>>(ei, eattr, asrc, adst, coef, meanv, amax,
                                                       asum, E, N);
    int scatter_blocks = (int)(((long long)TOT * 64 + 255) / 256);
    edge_scatter<<<scatter_blocks, 256, 0, stream>>>(ei, eattr, asrc, adst, coef, meanv, amax,
                                                     asum, hbuf, accum, E, N);
    finalize<<<(N * 64 + 255) / 256, 256, 0, stream>>>(accum, bias, x, out, N);
}